// KNearestNeighbour_66262755442915
// MI455X (gfx1250) — compile-verified
//
#include <hip/hip_runtime.h>

// ---------------------------------------------------------------------------
// kNN for MI455X (gfx1250): exact-f32 WMMA distance GEMM + register top-k.
// Each wave computes two 16x16 N-tiles per A-fragment (dual accumulators to
// pipeline v_wmma_f32_16x16x4_f32 and halve A-loads per WMMA).
// ---------------------------------------------------------------------------

typedef float v2f __attribute__((ext_vector_type(2)));
typedef float v8f __attribute__((ext_vector_type(8)));

#define QN        1024
#define NPTS      50000
#define DIM       512
#define KNN       10
#define NSPLIT    25
#define COLS_PS   2000          // NPTS / NSPLIT
#define CHUNK     128           // 4 waves * 32 cols
#define NCHUNKS   16            // ceil(COLS_PS / CHUNK)
#define XPITCH    516           // 512 + 4 pad -> conflict-free A-fragment reads
#define BIGF      3.0e38f

// ---------------------------------------------------------------------------
// 1) Row squared-norms: one wave32 per row.
// ---------------------------------------------------------------------------
__global__ void rownorm_kernel(const float* __restrict__ src,
                               float* __restrict__ out, int nrows) {
    int wave = (int)((blockIdx.x * blockDim.x + threadIdx.x) >> 5);
    int lane = threadIdx.x & 31;
    if (wave >= nrows) return;
    const float* r = src + (size_t)wave * DIM;
    float acc = 0.0f;
    #pragma unroll 4
    for (int c = lane; c < DIM; c += 32) {
        float v = r[c];
        acc = fmaf(v, v, acc);
    }
    #pragma unroll
    for (int off = 16; off > 0; off >>= 1)
        acc += __shfl_down(acc, off, 32);
    if (lane == 0) out[wave] = acc;
}

// ---------------------------------------------------------------------------
// 2) Main kernel: per (16-query tile, N-split) workgroup, 4 waves.
// ---------------------------------------------------------------------------
__global__ __launch_bounds__(128)
void knn_part_kernel(const float* __restrict__ Xg,
                     const float* __restrict__ data,
                     const float* __restrict__ d2,
                     const float* __restrict__ x2,
                     float* __restrict__ partD,
                     int*   __restrict__ partI) {
    __shared__ float Xlds[16 * XPITCH];
    __shared__ float tile[4][16 * 17];

    const int tid   = threadIdx.x;
    const int wave  = tid >> 5;
    const int lane  = tid & 31;
    const int mtile = (int)blockIdx.x / NSPLIT;
    const int split = (int)blockIdx.x % NSPLIT;
    const int m0    = mtile * 16;

    // Stage the 16x512 query tile into LDS (coalesced).
    for (int i = tid; i < 16 * DIM; i += 128) {
        int r = i >> 9, c = i & (DIM - 1);
        Xlds[r * XPITCH + c] = Xg[(size_t)(m0 + r) * DIM + c];
    }
    __syncthreads();

    // A-fragment layout (16x4 f32): row = lane%16, k-pair = 2*(lane/16).
    const int  halfk = (lane >> 4) << 1;     // 0 or 2
    const int  lcol  = lane & 15;
    const int  rbase = (lane >> 4) << 3;     // first C row this lane holds
    const float* aPtr = &Xlds[(lane & 15) * XPITCH + halfk];

    // Loop-invariant query norms for this lane's 8 C rows.
    float xs[8];
    #pragma unroll
    for (int j = 0; j < 8; ++j) xs[j] = x2[m0 + rbase + j];

    // Per-query (lane 0..15) sorted-ascending top-10 in registers.
    float td[KNN];
    int   ti[KNN];
    #pragma unroll
    for (int j = 0; j < KNN; ++j) { td[j] = BIGF; ti[j] = 0; }

    for (int ch = 0; ch < NCHUNKS; ++ch) {
        const int  lc0 = ch * CHUNK + wave * 32 + lcol;   // col within split
        const int  lc1 = lc0 + 16;
        const bool v0  = lc0 < COLS_PS;
        const bool v1  = lc1 < COLS_PS;
        const int  c0  = split * COLS_PS + (v0 ? lc0 : (COLS_PS - 1));
        const int  c1  = split * COLS_PS + (v1 ? lc1 : (COLS_PS - 1));
        const float* bPtr0 = data + (size_t)c0 * DIM + halfk;
        const float* bPtr1 = data + (size_t)c1 * DIM + halfk;

        // Prefetch next chunk's B rows (gfx1250 global_prefetch_b8).
        __builtin_prefetch(bPtr0 + (size_t)CHUNK * DIM, 0, 1);

        v8f acc0 = {};
        v8f acc1 = {};
        #pragma unroll 4
        for (int k0 = 0; k0 < DIM; k0 += 4) {
            v2f a  = *(const v2f*)(aPtr + k0);    // ds_load_b64, shared A
            v2f b0 = *(const v2f*)(bPtr0 + k0);   // global_load_b64 (L2)
            v2f b1 = *(const v2f*)(bPtr1 + k0);
            acc0 = __builtin_amdgcn_wmma_f32_16x16x4_f32(
                false, a, false, b0, (short)0, acc0, false, false);
            acc1 = __builtin_amdgcn_wmma_f32_16x16x4_f32(
                false, a, false, b1, (short)0, acc1, false, false);
        }

        const float dc0 = v0 ? d2[c0] : BIGF;   // BIGF self-poisons the tile
        const float dc1 = v1 ? d2[c1] : BIGF;
        const int   idbase = split * COLS_PS + ch * CHUNK + wave * 32;

        // ---- tile 0: transpose through LDS, then per-query insertion ----
        #pragma unroll
        for (int j = 0; j < 8; ++j)
            tile[wave][(rbase + j) * 17 + lcol] =
                fmaxf(xs[j] + dc0 - 2.0f * acc0[j], 0.0f);
        if (lane < 16) {
            #pragma unroll
            for (int c = 0; c < 16; ++c) {
                float dv = tile[wave][lane * 17 + c];
                if (dv < td[KNN - 1]) {
                    float cd = dv; int ci = idbase + c;
                    #pragma unroll
                    for (int j = 0; j < KNN; ++j) {
                        bool  sw = cd < td[j];
                        float od = td[j]; int oi = ti[j];
                        td[j] = sw ? cd : od;  ti[j] = sw ? ci : oi;
                        cd    = sw ? od : cd;  ci    = sw ? oi : ci;
                    }
                }
            }
        }

        // ---- tile 1 (DS ops from one wave stay in order; safe LDS reuse) ----
        #pragma unroll
        for (int j = 0; j < 8; ++j)
            tile[wave][(rbase + j) * 17 + lcol] =
                fmaxf(xs[j] + dc1 - 2.0f * acc1[j], 0.0f);
        if (lane < 16) {
            #pragma unroll
            for (int c = 0; c < 16; ++c) {
                float dv = tile[wave][lane * 17 + c];
                if (dv < td[KNN - 1]) {
                    float cd = dv; int ci = idbase + 16 + c;
                    #pragma unroll
                    for (int j = 0; j < KNN; ++j) {
                        bool  sw = cd < td[j];
                        float od = td[j]; int oi = ti[j];
                        td[j] = sw ? cd : od;  ti[j] = sw ? ci : oi;
                        cd    = sw ? od : cd;  ci    = sw ? oi : ci;
                    }
                }
            }
        }
    }

    // Emit this wave's partial list: disjoint column ranges per (split, wave).
    if (lane < 16) {
        size_t base = ((size_t)(m0 + lane) * NSPLIT + split) * (4 * KNN)
                      + (size_t)wave * KNN;
        #pragma unroll
        for (int j = 0; j < KNN; ++j) {
            partD[base + j] = td[j];
            partI[base + j] = ti[j];
        }
    }
}

// ---------------------------------------------------------------------------
// 3) Final merge + mode: one thread per query over 25*4*10 = 1000 entries.
// ---------------------------------------------------------------------------
__global__ void knn_final_kernel(const float* __restrict__ partD,
                                 const int*   __restrict__ partI,
                                 const int*   __restrict__ targets,
                                 float* __restrict__ pred) {
    int q = blockIdx.x * blockDim.x + threadIdx.x;
    if (q >= QN) return;

    float td[KNN];
    int   ti[KNN];
    #pragma unroll
    for (int j = 0; j < KNN; ++j) { td[j] = BIGF; ti[j] = 0; }

    const float* pd = partD + (size_t)q * (NSPLIT * 4 * KNN);
    const int*   pi = partI + (size_t)q * (NSPLIT * 4 * KNN);

    for (int e = 0; e < NSPLIT * 4 * KNN; ++e) {
        float dv = pd[e];
        if (dv < td[KNN - 1]) {
            float cd = dv; int ci = pi[e];
            #pragma unroll
            for (int j = 0; j < KNN; ++j) {
                bool  sw = cd < td[j];
                float od = td[j]; int oi = ti[j];
                td[j] = sw ? cd : od;  ti[j] = sw ? ci : oi;
                cd    = sw ? od : cd;  ci    = sw ? oi : ci;
            }
        }
    }

    int lab[KNN];
    #pragma unroll
    for (int j = 0; j < KNN; ++j) lab[j] = targets[ti[j]];

    // torch.mode: max count, smallest label wins ties.
    int bestLab = 0x7fffffff, bestCnt = 0;
    #pragma unroll
    for (int i = 0; i < KNN; ++i) {
        int c = 0;
        #pragma unroll
        for (int j = 0; j < KNN; ++j) c += (lab[j] == lab[i]) ? 1 : 0;
        if (c > bestCnt || (c == bestCnt && lab[i] < bestLab)) {
            bestCnt = c; bestLab = lab[i];
        }
    }
    pred[q] = (float)bestLab;
}

// ---------------------------------------------------------------------------
// Launch
// ---------------------------------------------------------------------------
extern "C" void kernel_launch(void* const* d_in, const int* in_sizes, int n_in,
                              void* d_out, int out_size, void* d_ws, size_t ws_size,
                              hipStream_t stream) {
    const float* Xg      = (const float*)d_in[0];
    const float* data    = (const float*)d_in[1];
    const int*   targets = (const int*)d_in[2];
    float*       pred    = (float*)d_out;

    // Workspace layout (floats):
    //   [0, NPTS)            d2
    //   [NPTS, NPTS+QN)      x2
    //   [51200, +QN*1000)    partial dists
    //   then                 partial idx (ints)
    float* ws    = (float*)d_ws;
    float* d2    = ws;
    float* x2    = ws + NPTS;
    float* partD = ws + 51200;
    int*   partI = (int*)(ws + 51200 + (size_t)QN * NSPLIT * 4 * KNN);

    // Squared norms (8 wave32 rows per 256-thread block).
    rownorm_kernel<<<(NPTS + 7) / 8, 256, 0, stream>>>(data, d2, NPTS);
    rownorm_kernel<<<(QN + 7) / 8,  256, 0, stream>>>(Xg,   x2, QN);

    // Distance GEMM + per-wave top-10: 64 query tiles x 25 column splits.
    knn_part_kernel<<<(QN / 16) * NSPLIT, 128, 0, stream>>>(
        Xg, data, d2, x2, partD, partI);

    // Merge + mode.
    knn_final_kernel<<<(QN + 127) / 128, 128, 0, stream>>>(
        partD, partI, targets, pred);

    (void)in_sizes; (void)n_in; (void)out_size; (void)ws_size;
}